// LQVitModel_85873576116614
// MI455X (gfx1250) — compile-verified
//
#include <hip/hip_runtime.h>
#include <hip/hip_bf16.h>

// FSQ forward for z[64,6,256,256] f32.
// d_out = [quantized out: 64*6*256*256 floats][indices: 64*256*256 floats]
//
// Memory-bound (~218 MB total traffic -> ~9.4us at 23.3 TB/s). No matrix
// contraction exists, so no WMMA; the gfx1250-specific paths used are:
//   - global_load_async_to_lds_b128 (ASYNCcnt) to stage all 6 strided
//     channel-plane segments with one wait, per-wave producer==consumer
//     so s_wait_asynccnt is the only sync needed
//   - v_tanh_f32 hardware transcendental (1 TRANS op vs ~30 VALU for the
//     OCML software tanh), keeping the kernel under the HBM roofline
//   - full-width b128 loads/stores for every byte of traffic

#define THREADS        256
#define PIX_PER_BLOCK  1024          // 4 pixels per thread
#define HW             65536         // 256*256
#define NCH            6
#define NB             64
#define OUT_ELEMS      (NB * NCH * HW)   // 25165824 floats, then indices

#if __has_builtin(__builtin_amdgcn_tanhf)
#define FAST_TANH(x) __builtin_amdgcn_tanhf(x)
#else
#define FAST_TANH(x) tanhf(x)
#endif

__global__ __launch_bounds__(THREADS)
void fsq_fwd_kernel(const float* __restrict__ z, float* __restrict__ out) {
  __shared__ float smem[NCH * PIX_PER_BLOCK];   // 24 KB / block

  const int tid  = threadIdx.x;
  const int blk  = blockIdx.x;
  const int b    = blk >> 6;                    // 64 blocks per image
  const int pix0 = (blk & 63) * PIX_PER_BLOCK;  // hw offset within image

  // Low 32 bits of a generic pointer into LDS == byte offset in the
  // workgroup's LDS allocation (aperture lives in the high word).
  const uint32_t ldsBase = (uint32_t)(uintptr_t)(&smem[0]);

  // Stage the 6 channel segments (1024 floats each) with async b128 copies.
  // Six ops outstanding on ASYNCcnt, single wait. Each lane moves 16 bytes.
#pragma unroll
  for (int c = 0; c < NCH; ++c) {
    uint32_t goff = (uint32_t)(((uint32_t)(b * NCH + c) * HW +
                                (uint32_t)(pix0 + tid * 4)) * 4u);
    uint32_t loff = ldsBase + (uint32_t)((c * PIX_PER_BLOCK + tid * 4) * 4);
    asm volatile("global_load_async_to_lds_b128 %0, %1, %2"
                 :: "v"(loff), "v"(goff), "s"(z)
                 : "memory");
  }
  // Wave consumes only its own staged data -> async counter is the only
  // ordering needed (no workgroup barrier).
  asm volatile("s_wait_asynccnt 0x0" ::: "memory");

  // Per-channel constants for levels [8,8,8,8,6,5], eps = 1e-3:
  //   half_l = (L-1)*(1+eps)/2 ; offset = 0.5 (L even) else 0
  //   shift  = atanh(offset/half_l)
  // Uniform codebook => nearest code k = clamp(round(zb*hw + hw), 0, L-1),
  // quantized value q = k/hw - 1, packed index = sum k_c * basis_c
  // with basis = cumprod-based [1,8,64,512,4096,24576].
  const float half_l[NCH] = {3.5035f, 3.5035f, 3.5035f, 3.5035f, 2.5025f, 2.002f};
  const float shiftc[NCH] = {0.14369536f, 0.14369536f, 0.14369536f, 0.14369536f,
                             0.20252444f, 0.0f};
  const float offs[NCH]   = {0.5f, 0.5f, 0.5f, 0.5f, 0.5f, 0.0f};
  const float hwq[NCH]    = {4.f, 4.f, 4.f, 4.f, 3.f, 2.f};
  const float rcp_hw[NCH] = {0.25f, 0.25f, 0.25f, 0.25f, 0.33333334f, 0.5f};
  const float lm1[NCH]    = {7.f, 7.f, 7.f, 7.f, 5.f, 4.f};
  const int   basis[NCH]  = {1, 8, 64, 512, 4096, 24576};

  int i0 = 0, i1 = 0, i2 = 0, i3 = 0;

#pragma unroll
  for (int c = 0; c < NCH; ++c) {
    float4 zv = *reinterpret_cast<const float4*>(&smem[c * PIX_PER_BLOCK + tid * 4]);
    float4 q;
    float k;

    k = fminf(fmaxf(rintf(fmaf(FAST_TANH(zv.x + shiftc[c]) * half_l[c] - offs[c],
                               hwq[c], hwq[c])), 0.f), lm1[c]);
    q.x = fmaf(k, rcp_hw[c], -1.f);  i0 += (int)k * basis[c];

    k = fminf(fmaxf(rintf(fmaf(FAST_TANH(zv.y + shiftc[c]) * half_l[c] - offs[c],
                               hwq[c], hwq[c])), 0.f), lm1[c]);
    q.y = fmaf(k, rcp_hw[c], -1.f);  i1 += (int)k * basis[c];

    k = fminf(fmaxf(rintf(fmaf(FAST_TANH(zv.z + shiftc[c]) * half_l[c] - offs[c],
                               hwq[c], hwq[c])), 0.f), lm1[c]);
    q.z = fmaf(k, rcp_hw[c], -1.f);  i2 += (int)k * basis[c];

    k = fminf(fmaxf(rintf(fmaf(FAST_TANH(zv.w + shiftc[c]) * half_l[c] - offs[c],
                               hwq[c], hwq[c])), 0.f), lm1[c]);
    q.w = fmaf(k, rcp_hw[c], -1.f);  i3 += (int)k * basis[c];

    *reinterpret_cast<float4*>(
        &out[(size_t)(b * NCH + c) * HW + (size_t)(pix0 + tid * 4)]) = q;
  }

  // Packed codes (max 122879, exactly representable in f32).
  float4 fi = make_float4((float)i0, (float)i1, (float)i2, (float)i3);
  *reinterpret_cast<float4*>(
      &out[(size_t)OUT_ELEMS + (size_t)b * HW + (size_t)(pix0 + tid * 4)]) = fi;
}

extern "C" void kernel_launch(void* const* d_in, const int* in_sizes, int n_in,
                              void* d_out, int out_size, void* d_ws, size_t ws_size,
                              hipStream_t stream) {
  (void)in_sizes; (void)n_in; (void)d_ws; (void)ws_size; (void)out_size;
  const float* z = (const float*)d_in[0];   // d_in[1..6] = codebook values,
                                            // reconstructed analytically.
  float* out = (float*)d_out;
  // 64 images * 65536 pixels / 1024 pixels-per-block = 4096 blocks
  fsq_fwd_kernel<<<dim3(4096), dim3(THREADS), 0, stream>>>(z, out);
}